// local_context_aggregation_75333726372151
// MI455X (gfx1250) — compile-verified
//
#include <hip/hip_runtime.h>
#include <math.h>
#include <stdint.h>

typedef float v2f __attribute__((ext_vector_type(2)));
typedef float v8f __attribute__((ext_vector_type(8)));
typedef unsigned int v4u __attribute__((ext_vector_type(4)));
typedef int v8i __attribute__((ext_vector_type(8)));
typedef int v4i __attribute__((ext_vector_type(4)));

#define NB   4
#define NF   32
#define NC   96     // F*3
#define NN   4096
#define KNN  20
#define CO   32     // output channels
#define KD   64     // GEMM K (concat edge feature)
#define EITER 8     // (b,n) points per edge_gemm block

static __device__ __forceinline__ v8f wmma_f32_4(v2f a, v2f b, v8f c) {
    return __builtin_amdgcn_wmma_f32_16x16x4_f32(false, a, false, b, (short)0, c, false, false);
}

// Issue a TDM 2D tile load: 96 rows x 16 cols of f32, row stride 4096 elems,
// packed row-major into LDS at ldsaddr. Wave-scalar; EXEC ignored by TDM.
static __device__ __forceinline__ void tdm_load_tile(const float* gaddr, unsigned ldsaddr) {
    uint64_t ga = (uint64_t)(uintptr_t)gaddr;
    unsigned galo = (unsigned)__builtin_amdgcn_readfirstlane((int)(ga & 0xFFFFFFFFu));
    unsigned gahi = (unsigned)__builtin_amdgcn_readfirstlane((int)(ga >> 32));
    unsigned lw   = (unsigned)__builtin_amdgcn_readfirstlane((int)ldsaddr);
    v4u g0;
    g0[0] = 1u;                                   // count=1, user descriptor
    g0[1] = lw;                                   // lds_addr
    g0[2] = galo;                                 // global_addr[31:0]
    g0[3] = (gahi & 0x1FFFFFFu) | (2u << 30);     // global_addr[56:32] | type=2
    v8i g1;
    g1[0] = (int)0x20000u;                        // data_size=4B
    g1[1] = (int)(4096u << 16);                   // tensor_dim0=4096 (lo16)
    g1[2] = (int)(96u << 16);                     // tensor_dim1=96  (lo16)
    g1[3] = (int)(16u << 16);                     // tile_dim0=16
    g1[4] = (int)96u;                             // tile_dim1=96, tile_dim2=0
    g1[5] = (int)4096u;                           // tensor_dim0_stride[31:0]
    g1[6] = 0;
    g1[7] = 0;
    v4i z4 = {0, 0, 0, 0};
    v8i z8 = {0, 0, 0, 0, 0, 0, 0, 0};
    __builtin_amdgcn_tensor_load_to_lds(g0, g1, z4, z4, z8, 0);
}

// ---------------------------------------------------------------------------
// Kernel 0: xx[b][n] = sum_c xf^2 ; zero the 64 stat accumulators
// ---------------------------------------------------------------------------
__global__ __launch_bounds__(256) void compute_xx(const float* __restrict__ x,
                                                  float* __restrict__ xx,
                                                  float* __restrict__ stats) {
    int g = blockIdx.x * 256 + threadIdx.x;
    if (g < 64) stats[g] = 0.0f;
    if (g >= NB * NN) return;
    int b = g / NN, n = g % NN;
    const float* xb = x + (size_t)b * NC * NN;
    float s = 0.0f;
    for (int c = 0; c < NC; ++c) {
        float v = xb[c * NN + n];
        s += v * v;
    }
    xx[g] = s;
}

// ---------------------------------------------------------------------------
// Kernel 1: kNN top-20. WMMA f32 16x16x4 Gram tiles; B-tiles streamed into
// LDS by the Tensor Data Mover, double-buffered so the DMA of tile t+1
// overlaps WMMA + top-k of tile t. Block = 128 threads (4 waves), each wave
// owns 16 query rows.
// ---------------------------------------------------------------------------
__global__ __launch_bounds__(128) void knn_topk(const float* __restrict__ x,
                                                const float* __restrict__ xx,
                                                int* __restrict__ idxout) {
    const int b    = blockIdx.x >> 6;     // 64 row-blocks per batch
    const int rb   = blockIdx.x & 63;
    const int wave = threadIdx.x >> 5;
    const int lane = threadIdx.x & 31;
    const int half = lane >> 4;
    const int l15  = lane & 15;
    const int n0   = rb * 64 + wave * 16;

    const float* xb  = x + (size_t)b * NC * NN;
    const float* xxb = xx + b * NN;

    __shared__ float stage[2][NC * 16];     // 12 KB double-buffered B tile
    __shared__ float scr[4][16][16];        // per-wave result tile
    __shared__ float topv[64][KNN];
    __shared__ int   topi[64][KNN];

    for (int i = threadIdx.x; i < 64 * KNN; i += 128) {
        topv[i / KNN][i % KNN] = -3.0e38f;
        topi[i / KNN][i % KNN] = 0;
    }
    // touch the stage buffer once so it has visible stores (TDM writes are
    // opaque to alias analysis); also warms LDS banks.
    for (int i = threadIdx.x; i < 2 * NC * 16; i += 128) stage[0][i] = 0.0f;

    // A fragments (loop invariant): lane l holds M=l&15, VGPR j -> K=4s+j+2*half
    v2f afr[24];
    for (int s = 0; s < 24; ++s) {
        int k0 = 4 * s + 2 * half;
        afr[s].x = xb[(k0 + 0) * NN + n0 + l15];
        afr[s].y = xb[(k0 + 1) * NN + n0 + l15];
    }
    float xxr[8];
    for (int r = 0; r < 8; ++r) xxr[r] = xxb[n0 + r + 8 * half];

    const unsigned lds0 = (unsigned)(uintptr_t)&stage[0][0];
    const unsigned lds1 = (unsigned)(uintptr_t)&stage[1][0];

    float kth = -3.0e38f;   // register cache of tv[KNN-1] (lanes 0-15)

    __syncthreads();
    if (wave == 0) tdm_load_tile(xb, lds0);   // prefetch tile 0 -> buf 0

    for (int t = 0; t < NN / 16; ++t) {
        if (wave == 0) {
            if (t + 1 < NN / 16) {
                tdm_load_tile(xb + (t + 1) * 16, ((t + 1) & 1) ? lds1 : lds0);
                __builtin_amdgcn_s_wait_tensorcnt((short)1);  // tile t landed
            } else {
                __builtin_amdgcn_s_wait_tensorcnt((short)0);
            }
        }
        asm volatile("" ::: "memory");   // TDM wrote LDS behind the compiler's back
        __syncthreads();                 // stage[t&1] visible to all waves

        const float* st = &stage[t & 1][0];
        v8f acc = {};
        for (int s = 0; s < 24; ++s) {
            v2f bfr;
            int k0 = 4 * s + 2 * half;
            bfr.x = st[(k0 + 0) * 16 + l15];
            bfr.y = st[(k0 + 1) * 16 + l15];
            acc = wmma_f32_4(afr[s], bfr, acc);
        }

        const int m0  = t * 16;
        float     xxc = xxb[m0 + l15];
        for (int r = 0; r < 8; ++r)
            scr[wave][r + 8 * half][l15] = 2.0f * acc[r] - xxr[r] - xxc;
        __syncthreads();                 // scr ready; all frag reads of buf done

        if (lane < 16) {
            float* tv = topv[wave * 16 + lane];
            int*   ti = topi[wave * 16 + lane];
            for (int j = 0; j < 16; ++j) {
                float v = scr[wave][lane][j];
                if (v > kth) {
                    int pos = KNN - 1;
                    while (pos > 0 && tv[pos - 1] < v) {
                        tv[pos] = tv[pos - 1];
                        ti[pos] = ti[pos - 1];
                        --pos;
                    }
                    tv[pos] = v;
                    ti[pos] = m0 + j;
                    kth = tv[KNN - 1];
                }
            }
        }
    }
    __syncthreads();
    for (int i = threadIdx.x; i < 64 * KNN; i += 128) {
        int r = i / KNN, kk = i % KNN;
        idxout[((size_t)b * NN + rb * 64 + r) * KNN + kk] = topi[r][kk];
    }
}

// ---------------------------------------------------------------------------
// Kernels 2/4: fused gather + edge-feature + WMMA channel GEMM(s).
// Persistent block (128 threads, 4 waves) handles EITER (b,n) points:
// weights staged to LDS once, BN stats accumulated in LDS across points.
// FULL=false: p only -> norm stats.  FULL=true: full activation + mean over k.
// ---------------------------------------------------------------------------
template <bool FULL>
__global__ __launch_bounds__(128) void edge_gemm(const float* __restrict__ x,
                                                 const float* __restrict__ Wf,
                                                 const float* __restrict__ Wd,
                                                 const int* __restrict__ idx,
                                                 const float* __restrict__ ab,
                                                 float* __restrict__ stats,
                                                 float* __restrict__ out) {
    const int base = blockIdx.x * EITER;
    const int wave = threadIdx.x >> 5;
    const int lane = threadIdx.x & 31;
    const int half = lane >> 4;
    const int l15  = lane & 15;

    __shared__ float wlds[2][CO * KD];   // 16 KB
    __shared__ float cols[1 + KNN][NC];  // 8 KB (col 0 = center)
    __shared__ float pt[CO * 64];        // 8 KB
    __shared__ float dt[CO * 64];        // 8 KB (FULL only)
    __shared__ float fN[CO * KNN];       // FULL only
    __shared__ float dotN[CO * KNN];
    __shared__ float dsqN[CO * KNN];
    __shared__ float snorm[64];          // stats partials (accumulated over EITER)

    for (int i = threadIdx.x; i < CO * KD; i += 128) {
        wlds[0][i] = Wf[i];
        if (FULL) wlds[1][i] = Wd[i];
    }
    if (!FULL && threadIdx.x < 64) snorm[threadIdx.x] = 0.0f;

    const int g   = wave * 16 + l15;     // global GEMM column 0..63
    const int kk  = g / 3;
    const int dd  = g - 3 * kk;
    const bool ok = (kk < KNN);

    for (int it = 0; it < EITER; ++it) {
        const int bn = base + it;
        const int b  = bn / NN;
        const int n  = bn % NN;

        __syncthreads();   // previous iteration fully consumed cols/pt/dt
        for (int i = threadIdx.x; i < (1 + KNN) * NC; i += 128) {
            int col  = i / NC, c = i % NC;
            int srcn = (col == 0) ? n : idx[((size_t)b * NN + n) * KNN + (col - 1)];
            cols[col][c] = x[((size_t)b * NC + c) * NN + srcn];
        }
        __syncthreads();

        // B fragments: rows K<32 -> x_nbr - x_ctr ; K>=32 -> x_ctr.
        v2f bfr[16];
        for (int s = 0; s < 16; ++s) {
            int k0 = 4 * s + 2 * half;
            float v0 = 0.0f, v1 = 0.0f;
            if (ok) {
                int k1 = k0 + 1;
                v0 = (k0 < NF) ? (cols[1 + kk][k0 * 3 + dd] - cols[0][k0 * 3 + dd])
                               : cols[0][(k0 - NF) * 3 + dd];
                v1 = (k1 < NF) ? (cols[1 + kk][k1 * 3 + dd] - cols[0][k1 * 3 + dd])
                               : cols[0][(k1 - NF) * 3 + dd];
            }
            bfr[s].x = v0;
            bfr[s].y = v1;
        }

        for (int mat = 0; mat < (FULL ? 2 : 1); ++mat) {
            const float* wl = wlds[mat];
            float* tile = mat ? dt : pt;
            for (int mt = 0; mt < 2; ++mt) {
                v8f acc = {};
                for (int s = 0; s < 16; ++s) {
                    v2f a;
                    int o  = mt * 16 + l15;
                    int k0 = 4 * s + 2 * half;
                    a.x = wl[o * KD + k0];
                    a.y = wl[o * KD + k0 + 1];
                    acc = wmma_f32_4(a, bfr[s], acc);
                }
                for (int r = 0; r < 8; ++r)
                    tile[(mt * 16 + r + 8 * half) * 64 + g] = acc[r];
            }
        }
        __syncthreads();

        if (!FULL) {
            for (int i = threadIdx.x; i < CO * KNN; i += 128) {
                int o = i / KNN, k2 = i % KNN, c0 = k2 * 3;
                float p0 = pt[o * 64 + c0], p1 = pt[o * 64 + c0 + 1], p2 = pt[o * 64 + c0 + 2];
                float nr = sqrtf(p0 * p0 + p1 * p1 + p2 * p2) + 1e-6f;
                atomicAdd(&snorm[o], nr);
                atomicAdd(&snorm[32 + o], nr * nr);
            }
        } else {
            for (int i = threadIdx.x; i < CO * KNN; i += 128) {
                int o = i / KNN, k2 = i % KNN, c0 = k2 * 3;
                float p0 = pt[o * 64 + c0], p1 = pt[o * 64 + c0 + 1], p2 = pt[o * 64 + c0 + 2];
                float d0 = dt[o * 64 + c0], d1 = dt[o * 64 + c0 + 1], d2 = dt[o * 64 + c0 + 2];
                float nr = sqrtf(p0 * p0 + p1 * p1 + p2 * p2) + 1e-6f;
                float nb = ab[o] * nr + ab[32 + o];  // gamma*(nr-mean)*rsqrt+beta
                float f  = nb / nr;
                fN[i]    = f;
                dotN[i]  = f * (p0 * d0 + p1 * d1 + p2 * d2);
                dsqN[i]  = d0 * d0 + d1 * d1 + d2 * d2;
            }
            __syncthreads();
            if (threadIdx.x < CO * 3) {
                int o = threadIdx.x / 3, d2i = threadIdx.x % 3;
                float sum = 0.0f;
                for (int k2 = 0; k2 < KNN; ++k2) {
                    int i     = o * KNN + k2;
                    float p   = pt[o * 64 + k2 * 3 + d2i] * fN[i];
                    float dv  = dt[o * 64 + k2 * 3 + d2i];
                    float dot = dotN[i];
                    float neg = p - dot / (dsqN[i] + 1e-6f) * dv;
                    float sel = (dot >= 0.0f) ? p : neg;
                    sum += 0.2f * p + 0.8f * sel;
                }
                out[(((size_t)b * CO + o) * 3 + d2i) * NN + n] = sum * (1.0f / (float)KNN);
            }
        }
    }

    if (!FULL) {
        __syncthreads();
        if (threadIdx.x < 64) atomicAdd(&stats[threadIdx.x], snorm[threadIdx.x]);
    }
}

// ---------------------------------------------------------------------------
// Kernel 3: fold global stats into per-channel affine  norm_bn = a*norm + b
// ---------------------------------------------------------------------------
__global__ void finalize_stats(const float* __restrict__ stats,
                               const float* __restrict__ gamma,
                               const float* __restrict__ beta,
                               float* __restrict__ ab) {
    int o = threadIdx.x;
    if (o < CO) {
        const float cnt = (float)NB * (float)NN * (float)KNN;
        float mean = stats[o] / cnt;
        float var  = stats[32 + o] / cnt - mean * mean;
        float a    = gamma[o] * rsqrtf(var + 1e-5f);
        ab[o]      = a;
        ab[32 + o] = beta[o] - mean * a;
    }
}

extern "C" void kernel_launch(void* const* d_in, const int* in_sizes, int n_in,
                              void* d_out, int out_size, void* d_ws, size_t ws_size,
                              hipStream_t stream) {
    const float* x     = (const float*)d_in[0];
    const float* Wf    = (const float*)d_in[1];
    const float* Wd    = (const float*)d_in[2];
    const float* gamma = (const float*)d_in[3];
    const float* beta  = (const float*)d_in[4];
    float* out = (float*)d_out;

    char* ws = (char*)d_ws;
    int*   idx   = (int*)ws;                                   // 4*4096*20 ints
    float* xx    = (float*)(ws + 1310720);                     // 4*4096 floats
    float* stats = (float*)(ws + 1310720 + 65536);             // 64 floats
    float* ab    = (float*)(ws + 1310720 + 65536 + 256);       // 64 floats

    compute_xx<<<(NB * NN) / 256, 256, 0, stream>>>(x, xx, stats);
    knn_topk<<<NB * (NN / 64), 128, 0, stream>>>(x, xx, idx);
    edge_gemm<false><<<(NB * NN) / EITER, 128, 0, stream>>>(x, Wf, Wd, idx, nullptr, stats, nullptr);
    finalize_stats<<<1, 64, 0, stream>>>(stats, gamma, beta, ab);
    edge_gemm<true><<<(NB * NN) / EITER, 128, 0, stream>>>(x, Wf, Wd, idx, ab, stats, out);
}